// RDF_7550552507059
// MI455X (gfx1250) — compile-verified
//
#include <hip/hip_runtime.h>
#include <math.h>

// ---------------------------------------------------------------------------
// RDF on MI455X (gfx1250, wave32).
//   - 96KB of positions -> fully cache resident; compute bound.
//   - Triangular supertile grid: blocks with bj<bi exit; off-diagonal blocks
//     counted with weight 2 at flush (g(d_ij)==g(d_ji)) -> halves pair work.
//   - Staging: global_load_async_to_lds_b32 (ASYNCcnt) scatters AoS positions
//     directly into SoA LDS layout; s_wait_asynccnt 0 + barrier.
//   - V_WMMA_F32_16X16X4_F32 computes 16x16 tiles of unwrapped pair d^2 on the
//     matrix pipe (co-executes with VALU); VALU applies the per-dim minimum-
//     image correction and a +/-16-bin Gaussian window (exact vs f32
//     reference: expf underflows to 0 beyond ~14.4 kernel widths).
//   - Guard-padded per-wave LDS histograms: unconditional ds_add_f32 per bin
//     (no per-iteration bounds check / EXEC churn), block reduce, global
//     f32 atomics.
//   - Raw v_sqrt_f32 / v_exp_f32 via __builtin_amdgcn_* (skip IEEE fixups).
// ---------------------------------------------------------------------------

typedef __attribute__((ext_vector_type(2))) float v2f;
typedef __attribute__((ext_vector_type(8))) float v8f;

#define NATOMS 8192
#define NB     400
#define BOXF   25.0f
#define HALFF  12.5f
#define RSTOPF 12.5f

// Gaussian kernel centers: offsets = linspace(0, 12.5, 400) -> spacing 12.5/399
#define WKER   (RSTOPF / 399.0f)
#define INVW   (399.0f / RSTOPF)
// exp(coeff*x^2) == exp2(C2L*x^2), C2L = coeff * log2(e)
#define C2L    ((-0.5f / (WKER * WKER)) * 1.44269504088896340736f)
// Histogram bin edges: bins = linspace(0, 12.5, 401) -> spacing 12.5/400
#define WBIN   (RSTOPF / 400.0f)

#define SUPER  128          // supertile edge (atoms) per block
#define NWAVE  8            // 256 threads = 8 wave32
#define PAD    16           // guard bins each side of the window
#define NBP    (NB + 48)    // padded per-wave histogram: idx = b + PAD, max 447

// LDS byte offset of a generic pointer that points into LDS.
#define LDS_OFF(p) ((unsigned)(unsigned long long)(__attribute__((address_space(3))) const void*)(p))

__global__ void rdf_zero(float* __restrict__ g) {
    int t = blockIdx.x * blockDim.x + threadIdx.x;
    if (t < NB) g[t] = 0.0f;
}

__global__ __launch_bounds__(256) void rdf_pairs(const float* __restrict__ pos,
                                                 float* __restrict__ ghist) {
    // Triangular grid: only bi <= bj blocks do work (weight 2 off-diagonal).
    if (blockIdx.y < blockIdx.x) return;

    __shared__ float cA[3][SUPER];            // SoA x/y/z, block's row atoms
    __shared__ float cB[3][SUPER];            // SoA x/y/z, block's col atoms
    __shared__ float n2A[SUPER], n2B[SUPER];  // |p|^2
    __shared__ float hist[NWAVE][NBP];        // guard-padded wave histograms

    const int tid  = threadIdx.x;
    const int wave = tid >> 5;
    const int lane = tid & 31;
    const int rowBase = blockIdx.x * SUPER;
    const int colBase = blockIdx.y * SUPER;
    const float flushScale = (blockIdx.x == blockIdx.y) ? 1.0f : 2.0f;

    // ---- async-stage positions into LDS, deswizzling AoS -> SoA ----
    // 2 * 128 atoms * 3 dwords = 768 dwords; 3 async b32 ops per thread.
    {
        const unsigned long long posBase = (unsigned long long)pos;
#pragma unroll
        for (int q = 0; q < 3; ++q) {
            const int g    = q * 256 + tid;        // 0..767
            const int grp  = (g >= 384) ? 1 : 0;   // 0 = rows (A), 1 = cols (B)
            const int gl   = g - 384 * grp;        // 0..383
            const int atom = gl / 3, dim = gl % 3;
            const int gatom = (grp ? colBase : rowBase) + atom;
            const unsigned long long gaddr =
                posBase + (unsigned long long)(3 * gatom + dim) * 4ull;
            float* dst = grp ? &cB[dim][atom] : &cA[dim][atom];
            const unsigned ldsoff = LDS_OFF(dst);
            asm volatile("global_load_async_to_lds_b32 %0, %1, off"
                         :: "v"(ldsoff), "v"(gaddr) : "memory");
        }
        asm volatile("s_wait_asynccnt 0" ::: "memory");
    }
    for (int h = tid; h < NWAVE * NBP; h += 256) (&hist[0][0])[h] = 0.0f;
    __syncthreads();

    if (tid < SUPER) {
        n2A[tid] = cA[0][tid] * cA[0][tid] + cA[1][tid] * cA[1][tid]
                 + cA[2][tid] * cA[2][tid];
    } else {
        const int t2 = tid - SUPER;
        n2B[t2] = cB[0][t2] * cB[0][t2] + cB[1][t2] * cB[1][t2]
                + cB[2][t2] * cB[2][t2];
    }
    __syncthreads();

    // ---- per-wave 16-row strip; loop over eight 16x16 column tiles ----
    const int m  = lane & 15;     // row (A) / col (B,C,D) index within tile
    const int hi = lane >> 4;     // half-wave selector
    const int i0 = wave * 16;     // row offset of this wave's strip

    // A fragment (16x4, f32): lanes 0-15 hold K0/K1, lanes 16-31 hold K2/K3.
    // A_i = (x_i, y_i, z_i, 1)
    v2f afrag;
    afrag.x = hi ? cA[2][i0 + m] : cA[0][i0 + m];
    afrag.y = hi ? 1.0f          : cA[1][i0 + m];

    // Row data for the min-image correction: C/D VGPR r <-> row i0 + r + 8*hi
    float pix[8], piy[8], piz[8];
    v8f cfrag;
#pragma unroll
    for (int r = 0; r < 8; ++r) {
        const int ii = i0 + r + 8 * hi;
        pix[r] = cA[0][ii]; piy[r] = cA[1][ii]; piz[r] = cA[2][ii];
        cfrag[r] = n2A[ii];                       // C preloaded with |p_i|^2
    }

    for (int jt = 0; jt < SUPER / 16; ++jt) {
        const int jn = jt * 16 + m;               // B column within supertile
        const float xj = cB[0][jn], yj = cB[1][jn], zj = cB[2][jn];

        // B fragment (4x16, f32): lanes 0-15 hold K0/K1, lanes 16-31 K2/K3.
        // B_j = (-2x_j, -2y_j, -2z_j, |p_j|^2)
        v2f bfrag;
        bfrag.x = hi ? (-2.0f * zj) : (-2.0f * xj);
        bfrag.y = hi ? n2B[jn]      : (-2.0f * yj);

        // d2[r] = |p_i|^2 + |p_j|^2 - 2 p_i . p_j   (unwrapped), one tile/op
        v8f d2 = __builtin_amdgcn_wmma_f32_16x16x4_f32(
            /*neg_a=*/false, afrag, /*neg_b=*/false, bfrag,
            /*c_mod=*/(short)0, cfrag, /*reuse_a=*/false, /*reuse_b=*/false);

        const int jglob = colBase + jn;
#pragma unroll
        for (int r = 0; r < 8; ++r) {
            const int iglob = rowBase + i0 + r + 8 * hi;
            if (iglob == jglob) continue;         // reference masks dist > 0

            float dsum = d2[r];
            // per-dimension minimum-image correction: d2 += box^2 - 2*box*|dx|
            float dx = fabsf(pix[r] - xj);
            if (dx > HALFF) dsum = fmaf(-2.0f * BOXF, dx, dsum + BOXF * BOXF);
            float dy = fabsf(piy[r] - yj);
            if (dy > HALFF) dsum = fmaf(-2.0f * BOXF, dy, dsum + BOXF * BOXF);
            float dz = fabsf(piz[r] - zj);
            if (dz > HALFF) dsum = fmaf(-2.0f * BOXF, dz, dsum + BOXF * BOXF);

            // raw v_sqrt_f32: ~1 ulp, skip the IEEE fixup sequence
            const float d = __builtin_amdgcn_sqrtf(fmaxf(dsum, 0.0f));
            if (d <= 0.0f) continue;

            // +/-16 kernel-width window; everything outside underflows to 0.0f
            const int bc  = (int)floorf(d * INVW);
            const int blo = bc - 16;
            if (blo >= NB) continue;              // ~48% of pairs exit here

            float* hrow = &hist[wave][blo + PAD]; // guard pad: index >= 0
            const float dd0 = d - (float)blo * WKER;
#pragma unroll
            for (int t = 0; t < 33; ++t) {
                const float dd = dd0 - (float)t * WKER;   // constant subtract
                const float g = __builtin_amdgcn_exp2f(C2L * (dd * dd));
                atomicAdd(hrow + t, g);                   // ds_add_f32
            }
        }
    }

    // ---- reduce 8 wave-private histograms, flush real bins to global ----
    __syncthreads();
    for (int b = tid; b < NB; b += 256) {
        float s = 0.0f;
#pragma unroll
        for (int wv = 0; wv < NWAVE; ++wv) s += hist[wv][b + PAD];
        atomicAdd(&ghist[b], s * flushScale);
    }
}

__global__ __launch_bounds__(512) void rdf_finalize(const float* __restrict__ ghist,
                                                    float* __restrict__ out) {
    __shared__ float sh[512];
    const int t = threadIdx.x;
    const float v = (t < NB) ? ghist[t] : 0.0f;
    sh[t] = v;
    __syncthreads();
    for (int s = 256; s > 0; s >>= 1) {
        if (t < s) sh[t] += sh[t + s];
        __syncthreads();
    }
    const float total = sh[0];

    if (t < NB) {
        const float count = v / total;
        out[t] = count;                                   // count[0..399]
        const float b0 = (float)t * WBIN;
        const float b1 = (float)(t + 1) * WBIN;
        // rdf = count / (vol_bins / V); 4pi/3 cancels
        out[801 + t] = count * (RSTOPF * RSTOPF * RSTOPF)
                             / (b1 * b1 * b1 - b0 * b0 * b0);
    }
    if (t < NB + 1) out[400 + t] = (float)t * WBIN;       // bins[0..400]
}

extern "C" void kernel_launch(void* const* d_in, const int* in_sizes, int n_in,
                              void* d_out, int out_size, void* d_ws, size_t ws_size,
                              hipStream_t stream) {
    const float* pos = (const float*)d_in[0];   // [8192*3] f32
    float* out  = (float*)d_out;                // 1201 f32: count|bins|rdf
    float* hist = (float*)d_ws;                 // 400 f32 accumulator

    rdf_zero<<<dim3((NB + 255) / 256), dim3(256), 0, stream>>>(hist);

    dim3 grid(NATOMS / SUPER, NATOMS / SUPER);  // 64 x 64, upper triangle works
    rdf_pairs<<<grid, dim3(256), 0, stream>>>(pos, hist);

    rdf_finalize<<<dim3(1), dim3(512), 0, stream>>>(hist, out);
}